// TopKTopPSampler_44109314130527
// MI455X (gfx1250) — compile-verified
//
#include <hip/hip_runtime.h>
#include <math.h>
#include <stdint.h>

// ---------------------------------------------------------------------------
// Top-k / top-p sampling + softmax, B=128 rows x V=128000 vocab, fp32.
//
// Per row (one 1024-thread workgroup = 32 wave32 waves):
//   A) 12-bit-key histogram + row max; HBM->LDS staged with the CDNA5
//      Tensor Data Mover (tensor_load_to_lds + s_wait_tensorcnt),
//      double-buffered so the DMA overlaps the LDS-side histogram.
//   B) collect candidate values (bin >= cutoff bin) -> <=1024 candidates
//      (b128 global loads, L2-resident re-read).
//   C) bitonic-sort candidates descending; thresh = k-th largest value.
//   D) inclusive prefix-scan of candidate exp() -> exact Z1, top-p cutoff
//      v_cut (survive iff prefix-of-top-mass < p*Z1; max always kept), and
//      exact Z2 -- all from candidates only; no extra full-row passes.
//      (Everything >= v_cut is provably inside the candidate list.)
//   E) out = exp((x>=v_cut ? x : TINY) - M)/Z2, b128 loads + non-temporal
//      b128 stores (keep L2 for the logits, the output is write-once).
// Roofline: minimum traffic 131 MB ~ 5.6 us @ 23.3 TB/s; this does
// 3 reads + 1 write, only the first read from HBM (65.5 MB < 192 MB L2).
// ---------------------------------------------------------------------------

#define BLK    1024
#define NBINS  4096
#define NCAND  1024           // must equal BLK
#define TILE   4000           // floats per TDM tile (16 KB); 128000 % 4000 == 0
#define FP16_TINY 6.103515625e-05f

typedef __attribute__((ext_vector_type(4))) unsigned int tdm_v4u;
typedef __attribute__((ext_vector_type(8))) int          tdm_v8i;
typedef __attribute__((ext_vector_type(4))) int          tdm_v4i;
typedef __attribute__((ext_vector_type(4))) float        v4f;

#if defined(__HIP_DEVICE_COMPILE__) && __has_builtin(__builtin_amdgcn_tensor_load_to_lds) && __has_builtin(__builtin_amdgcn_s_wait_tensorcnt)
#define USE_TDM 1
#else
#define USE_TDM 0
#endif

// Monotone order-preserving map float -> u32 (handles sign).
__device__ __forceinline__ unsigned mapf(float x) {
  unsigned b = __float_as_uint(x);
  return (b & 0x80000000u) ? ~b : (b | 0x80000000u);
}

#if USE_TDM
// Issue a 1-D TDM load of `nelem` fp32 from global `gsrc` into LDS offset
// `ldsOff`. D# layout per CDNA5 ISA (count=1, type=2,
// data_size=4B, tile_dim0=nelem, tensor_dim0=nelem, no pad/iterate/multicast).
__device__ __forceinline__ void tdm_issue(const float* gsrc, unsigned ldsOff, int nelem) {
  unsigned long long ga = (unsigned long long)(uintptr_t)gsrc;
  tdm_v4u g0;
  g0[0] = 1u;                                                   // count=1, user desc
  g0[1] = ldsOff;                                               // lds_addr (bytes)
  g0[2] = (unsigned)ga;                                         // global_addr[31:0]
  g0[3] = ((unsigned)(ga >> 32) & 0x01FFFFFFu) | 0x80000000u;   // addr[56:32], type=2
  unsigned td0 = (unsigned)nelem;
  tdm_v8i g1;
  g1[0] = (int)(2u << 16);                                      // data_size=4B
  g1[1] = (int)((td0 & 0xFFFFu) << 16);                         // tensor_dim0 lo16
  g1[2] = (int)(((td0 >> 16) & 0xFFFFu) | (1u << 16));          // tensor_dim0 hi, tensor_dim1=1
  g1[3] = (int)((td0 & 0xFFFFu) << 16);                         // tile_dim0
  g1[4] = 0;                                                    // tile_dim1/2 unused
  g1[5] = (int)td0;                                             // tensor_dim0_stride lo
  g1[6] = 0;
  g1[7] = 0;
  tdm_v4i z4 = {0, 0, 0, 0};
#if __has_include(<hip/amd_detail/amd_gfx1250_TDM.h>)
  tdm_v8i z8 = {0, 0, 0, 0, 0, 0, 0, 0};
  __builtin_amdgcn_tensor_load_to_lds(g0, g1, z4, z4, z8, 0);   // 6-arg toolchain
#else
  __builtin_amdgcn_tensor_load_to_lds(g0, g1, z4, z4, 0);       // 5-arg ROCm 7.2
#endif
}
#endif

__global__ __launch_bounds__(BLK) void topk_topp_softmax_kernel(
    const float* __restrict__ logits, const int* __restrict__ kArr,
    const float* __restrict__ pArr, const int* __restrict__ flagK,
    const int* __restrict__ flagP, float* __restrict__ out, int V) {
  __shared__ __align__(16) float s_buf[2][TILE];
  __shared__ unsigned s_hist[NBINS];
  __shared__ float    s_cand[NCAND];
  __shared__ float    s_scan[NCAND];
  __shared__ float    s_red[BLK];
  __shared__ unsigned s_maxu;
  __shared__ int      s_cnt, s_bcut, s_cge, s_dlast, s_cvc;

  const int tid = threadIdx.x;
  const int row = blockIdx.x;
  const float* x = logits + (size_t)row * (size_t)V;
  float*       o = out    + (size_t)row * (size_t)V;

  const int   noK = flagK[0];
  const int   noP = flagP[0];
  const int   kr  = kArr[row];
  const float pr  = pArr[row];
  const int   K   = kr < 1 ? 1 : (kr > NCAND - 1 ? NCAND - 1 : kr);
  const bool  plain = (noK != 0) && (noP != 0);

  // ---- Phase A: histogram (top 12 bits of order-mapped key) + row max -----
  for (int i = tid; i < NBINS; i += BLK) s_hist[i] = 0u;
  if (tid == 0) s_maxu = 0u;
  __syncthreads();

  unsigned localMax = 0u;
#if USE_TDM
  {
    const int nt = V / TILE;                 // full TDM tiles
    if (tid < 32 && nt > 0)
      tdm_issue(x, (unsigned)(uintptr_t)&s_buf[0][0], TILE);
    for (int t = 0; t < nt; ++t) {
      if (tid < 32) {                        // wave 0 drives the DMA pipeline
        if (t + 1 < nt) {
          tdm_issue(x + (size_t)(t + 1) * TILE,
                    (unsigned)(uintptr_t)&s_buf[(t + 1) & 1][0], TILE);
          __builtin_amdgcn_s_wait_tensorcnt(1);   // tile t landed in LDS
        } else {
          __builtin_amdgcn_s_wait_tensorcnt(0);
        }
      }
      __syncthreads();                       // publish tile t to all waves
      const v4f* b4 = (const v4f*)&s_buf[t & 1][0];
      for (int j = tid; j < TILE / 4; j += BLK) {   // ds_load_b128
        v4f w = b4[j];
        unsigned u0 = mapf(w.x), u1 = mapf(w.y), u2 = mapf(w.z), u3 = mapf(w.w);
        atomicAdd(&s_hist[u0 >> 20], 1u);
        atomicAdd(&s_hist[u1 >> 20], 1u);
        atomicAdd(&s_hist[u2 >> 20], 1u);
        atomicAdd(&s_hist[u3 >> 20], 1u);
        unsigned m01 = u0 > u1 ? u0 : u1;
        unsigned m23 = u2 > u3 ? u2 : u3;
        unsigned m = m01 > m23 ? m01 : m23;
        localMax = localMax > m ? localMax : m;
      }
      __syncthreads();                       // done reading before buffer reuse
    }
    for (int j = nt * TILE + tid; j < V; j += BLK) {   // tail (none for V=128000)
      unsigned u = mapf(x[j]);
      atomicAdd(&s_hist[u >> 20], 1u);
      localMax = localMax > u ? localMax : u;
    }
  }
#else
  for (int j = tid; j < V; j += BLK) {
    unsigned u = mapf(x[j]);
    atomicAdd(&s_hist[u >> 20], 1u);
    localMax = localMax > u ? localMax : u;
  }
#endif
  atomicMax(&s_maxu, localMax);
  __syncthreads();

  const unsigned mu = s_maxu;
  const float M = __uint_as_float((mu & 0x80000000u) ? (mu & 0x7FFFFFFFu) : ~mu);

  // ---- cutoff bin: largest bin b with count(bin >= b) >= K ---------------
  if (tid == 0) {
    unsigned acc = 0; int bc = 0;
    for (int b2 = NBINS - 1; b2 >= 0; --b2) {
      acc += s_hist[b2];
      if (acc >= (unsigned)K) { bc = b2; break; }
    }
    s_bcut = bc;
    s_cnt  = 0;
  }
  __syncthreads();

  float thresh = -INFINITY;
  float vcut   = -INFINITY;
  float Mm     = M;
  float Z1 = 1.f, Z2 = 1.f;

  if (!plain) {
    // ---- Phase B: collect candidates (b128 loads, L2-resident) ------------
    const int bc = s_bcut;
    const v4f* xv = (const v4f*)x;
    const int V4 = V >> 2;
    for (int j = tid; j < V4; j += BLK) {
      v4f w = xv[j];
      #pragma unroll
      for (int c = 0; c < 4; ++c) {
        float v = (c == 0) ? w.x : (c == 1) ? w.y : (c == 2) ? w.z : w.w;
        if ((int)(mapf(v) >> 20) >= bc) {
          int pos = atomicAdd(&s_cnt, 1);
          if (pos < NCAND) s_cand[pos] = v;
        }
      }
    }
    for (int j = (V4 << 2) + tid; j < V; j += BLK) {
      float v = x[j];
      if ((int)(mapf(v) >> 20) >= bc) {
        int pos = atomicAdd(&s_cnt, 1);
        if (pos < NCAND) s_cand[pos] = v;
      }
    }
    __syncthreads();
    int n = s_cnt; if (n > NCAND) n = NCAND;
    if (tid >= n) s_cand[tid] = -INFINITY;   // NCAND == BLK
    __syncthreads();

    // ---- Phase C: bitonic sort descending over NCAND ----------------------
    for (unsigned ksz = 2; ksz <= (unsigned)NCAND; ksz <<= 1) {
      for (unsigned j2 = ksz >> 1; j2 > 0; j2 >>= 1) {
        unsigned i = (unsigned)tid, ixj = i ^ j2;
        if (ixj > i) {
          float a = s_cand[i], c = s_cand[ixj];
          bool descBlock = ((i & ksz) == 0);
          if (descBlock == (a < c)) { s_cand[i] = c; s_cand[ixj] = a; }
        }
        __syncthreads();
      }
    }
    thresh = noK ? -INFINITY : s_cand[K - 1];   // k-th largest (value-based mask)
    Mm = fmaxf(M, FP16_TINY);                   // max of masked array
  }

  // ---- full-row Z1 only for degenerate flag paths (never for flags==0) ----
  float Z1full = 0.f;
  if (plain || (noK != 0)) {
    float ls = 0.f;
    for (int j = tid; j < V; j += BLK) ls += expf(x[j] - Mm);  // thresh=-inf => y=x
    s_red[tid] = ls;
    __syncthreads();
    for (int st = BLK >> 1; st > 0; st >>= 1) {
      if (tid < st) s_red[tid] += s_red[tid + st];
      __syncthreads();
    }
    Z1full = s_red[0];
    __syncthreads();
  }

  if (!plain) {
    // ---- Phase D: candidate prefix-scan -> Z1, v_cut, Z2 ------------------
    const float eT = expf(FP16_TINY - Mm);
    const float v  = s_cand[tid];
    const bool kept = (v >= thresh) && (v > -INFINITY);
    s_scan[tid] = kept ? expf(v - Mm) : 0.f;
    if (tid == 0) { s_cge = -1; s_dlast = 0; s_cvc = 0; }
    __syncthreads();
    if (kept) atomicMax(&s_cge, tid);            // kept values are a prefix
    __syncthreads();
    const int cge = s_cge + 1;                   // #kept candidates (>=1)
    for (int off = 1; off < NCAND; off <<= 1) {  // Hillis-Steele inclusive scan
      float add = (tid >= off) ? s_scan[tid - off] : 0.f;
      __syncthreads();
      s_scan[tid] += add;
      __syncthreads();
    }
    const float keptSum = s_scan[cge - 1];
    // Exact Z1 from candidates (kept set is fully inside the candidate list);
    // TINY tail summed in closed form. Degenerate noK path uses the full sum.
    Z1 = (noK != 0) ? Z1full : ((float)(V - cge) * eT + keptSum);

    if (noP == 0) {
      // descending rank d survives iff sum of probs above it < p (this is the
      // reference's "cumsum <= 1-p" rule: the TINY tail sits below all kept
      // values in the ascending cumsum); d=0 (the max) always survives.
      const float pZ1 = pr * Z1;
      const float Pprev = (tid > 0) ? s_scan[tid - 1] : 0.f;
      if (tid < cge && (tid == 0 || Pprev < pZ1)) atomicMax(&s_dlast, tid);
      __syncthreads();
      vcut = s_cand[s_dlast];
      // every element >= vcut is a candidate (monotone binning, vcut is a
      // candidate) => Z2 is exact from the same scan.
      if (kept && v >= vcut) atomicMax(&s_cvc, tid);
      __syncthreads();
      const int cvc = s_cvc + 1;                 // #survivors
      Z2 = (float)(V - cvc) * eT + s_scan[cvc - 1];
    } else {
      vcut = thresh;                             // top-k only: z == y
      Z2   = Z1;
    }
  } else {
    vcut = -INFINITY;                            // plain softmax
    Z1 = Z2 = Z1full;
  }
  __syncthreads();

  // ---- Phase E: write softmax, b128 loads + non-temporal b128 stores ------
  const float invZ2   = 1.f / Z2;
  const float tinyOut = expf(FP16_TINY - Mm) * invZ2;
  const int V4 = V >> 2;
  const v4f* xv = (const v4f*)x;
  v4f*       ov = (v4f*)o;
  for (int j = tid; j < V4; j += BLK) {
    if (j + BLK < V4) __builtin_prefetch(&xv[j + BLK], 0, 3);  // global_prefetch
    v4f vv = xv[j];
    v4f r;
    r.x = (vv.x >= vcut) ? expf(vv.x - Mm) * invZ2 : tinyOut;
    r.y = (vv.y >= vcut) ? expf(vv.y - Mm) * invZ2 : tinyOut;
    r.z = (vv.z >= vcut) ? expf(vv.z - Mm) * invZ2 : tinyOut;
    r.w = (vv.w >= vcut) ? expf(vv.w - Mm) * invZ2 : tinyOut;
    __builtin_nontemporal_store(r, &ov[j]);      // write-once output
  }
  for (int j = (V4 << 2) + tid; j < V; j += BLK) {
    float v = x[j];
    o[j] = (v >= vcut) ? expf(v - Mm) * invZ2 : tinyOut;
  }
}

extern "C" void kernel_launch(void* const* d_in, const int* in_sizes, int n_in,
                              void* d_out, int out_size, void* d_ws, size_t ws_size,
                              hipStream_t stream) {
  (void)n_in; (void)out_size; (void)d_ws; (void)ws_size;
  const float* logits = (const float*)d_in[0];
  const int*   k      = (const int*)d_in[1];
  const float* p      = (const float*)d_in[2];
  const int*   noK    = (const int*)d_in[3];
  const int*   noP    = (const int*)d_in[4];
  float*       out    = (float*)d_out;
  const int B = in_sizes[1];              // k has one entry per row
  const int V = in_sizes[0] / B;
  topk_topp_softmax_kernel<<<B, BLK, 0, stream>>>(logits, k, p, noK, noP, out, V);
}